// MyNet_23699629539614
// MI455X (gfx1250) — compile-verified
//
#include <hip/hip_runtime.h>

// ============================================================================
// MPNN forward, algebraically collapsed.
//   out[g] = (Σ_{e: batch[dst_e]=g} [x[src_e], ea_e, 1]) · v + c
//   v = [W_msg; b_msg] @ (W1 @ W2)  (49 floats incl. c, kept in d_ws)
// Memory-bound: ~60-160 MB traffic @ 23.3 TB/s. The 48-wide per-edge dot is
// executed 16 edges at a time with V_WMMA_F32_16X16X4_F32 so the matrix pipe
// does the math while VMEM streams gathers.
// ============================================================================

typedef __attribute__((ext_vector_type(2))) float v2f;
typedef __attribute__((ext_vector_type(8))) float v8f;

#define N_NODES    200000
#define N_EDGES    800000
#define NUM_GRAPHS 4096
#define F_NODE     40
#define F_EDGE     7
#define INNER      512
#define NGRP       (N_EDGES / 16)   // 50000 groups of 16 edges

// ---------------------------------------------------------------------------
// Prep: v[0..46] = W_msg @ u, v[47] = b_msg·u (bias-feature), v[48] = c,
// where u = W1 @ W2.  One block, trivial cost.
// ---------------------------------------------------------------------------
__global__ __launch_bounds__(512)
void prep_kernel(const float* __restrict__ Wmsg, const float* __restrict__ bmsg,
                 const float* __restrict__ W1,   const float* __restrict__ b1,
                 const float* __restrict__ W2,   const float* __restrict__ b2,
                 float* __restrict__ vws) {
    __shared__ float u[INNER];
    int t = threadIdx.x;
    float s = 0.f;
    #pragma unroll
    for (int k = 0; k < 50; ++k) s += W1[t * 50 + k] * W2[k];
    u[t] = s;
    __syncthreads();
    if (t < 47) {
        float a = 0.f;
        for (int i = 0; i < INNER; ++i) a += Wmsg[t * INNER + i] * u[i];
        vws[t] = a;
    } else if (t == 47) {              // b_msg folded as constant-1 feature
        float a = 0.f;
        for (int i = 0; i < INNER; ++i) a += bmsg[i] * u[i];
        vws[47] = a;
    } else if (t == 48) {              // c = b1·W2 + b2
        float a = b2[0];
        for (int k = 0; k < 50; ++k) a += b1[k] * W2[k];
        vws[48] = a;
    }
}

// out[g] = c for every graph (also covers graphs with no incoming edges).
__global__ void init_kernel(float* __restrict__ out, const float* __restrict__ vws) {
    int i = blockIdx.x * blockDim.x + threadIdx.x;
    if (i < NUM_GRAPHS) out[i] = vws[48];
}

// ---------------------------------------------------------------------------
// Main edge kernel. Each wave handles 16 edges per iteration via WMMA:
//   A (16x4 f32): lane m / m+16 hold edge m's features [4c+2h, 4c+2h+1]
//                 (ISA 16x4 f32 A layout: lanes 0-15 K=0,1; lanes 16-31 K=2,3)
//   B (4x16 f32): v-chunk broadcast over all N columns (layout mirrors A),
//                 so D[m][n] = dot(edge m, v-chunk) for every n.
//   D accumulates over 12 chunks (40 x-features, 7 edge features, 1 bias).
// Graph ids are staged through a per-wave LDS buffer (same-wave LDS ops are
// in-order, no barrier needed); lanes 0/16 scatter the 16 dots into the
// per-block LDS accumulator in a single exec-masked region.
// ---------------------------------------------------------------------------
__global__ __launch_bounds__(256)
void edge_kernel(const float* __restrict__ x,    const float* __restrict__ ea,
                 const int*   __restrict__ eidx, const int*   __restrict__ batch,
                 const float* __restrict__ vws,  float* __restrict__ out) {
    __shared__ float acc[NUM_GRAPHS];        // 16 KB of the 320 KB WGP LDS
    __shared__ int   gbuf[8 * 16];           // per-wave graph-id staging
    const int tid = threadIdx.x;
    for (int i = tid; i < NUM_GRAPHS; i += 256) acc[i] = 0.f;
    __syncthreads();

    const int lane = tid & 31;
    const int half = lane >> 4;              // which K-half this lane supplies
    const int m    = lane & 15;              // edge-in-group == A row
    const int w    = tid >> 5;               // wave id within block

    // Per-lane B fragments: B[k][n] = v[4c+k] (broadcast over n).
    v2f B[12];
    #pragma unroll
    for (int c = 0; c < 12; ++c)
        B[c] = *(const v2f*)(vws + 4 * c + 2 * half);   // 8B aligned (even idx)

    const int wid   = blockIdx.x * (blockDim.x >> 5) + (tid >> 5);
    const int nwave = gridDim.x * (blockDim.x >> 5);

    for (int grp = wid; grp < NGRP; grp += nwave) {      // wave-uniform loop
        const int e   = grp * 16 + m;
        const int src = eidx[e];
        const int dst = eidx[N_EDGES + e];
        const int g   = batch[dst];
        const float* xrow = x  + (size_t)src * F_NODE;   // 160B rows, 16B aligned
        const float* erow = ea + (size_t)e   * F_EDGE;

        // Stage graph ids (lanes m and 16+m write identical values: benign).
        gbuf[w * 16 + m] = g;

        // Prefetch next group's linear streams (x gather is data-dependent).
        if (grp + nwave < NGRP) {                        // wave-uniform guard
            const int en = (grp + nwave) * 16 + m;
            __builtin_prefetch(eidx + en, 0, 1);
            __builtin_prefetch(eidx + N_EDGES + en, 0, 1);
            __builtin_prefetch(ea + (size_t)en * F_EDGE, 0, 1);
        }

        v8f d = {0.f, 0.f, 0.f, 0.f, 0.f, 0.f, 0.f, 0.f};
        // chunks 0..9: node features 0..39 (aligned float2 gathers, L2-resident)
        #pragma unroll
        for (int c = 0; c < 10; ++c) {
            v2f a = *(const v2f*)(xrow + 4 * c + 2 * half);
            d = __builtin_amdgcn_wmma_f32_16x16x4_f32(false, a, false, B[c],
                                                      (short)0, d, false, false);
        }
        // chunk 10: features 40..43 -> ea[0..3]
        v2f a10;
        a10.x = erow[0 + 2 * half];
        a10.y = erow[1 + 2 * half];
        d = __builtin_amdgcn_wmma_f32_16x16x4_f32(false, a10, false, B[10],
                                                  (short)0, d, false, false);
        // chunk 11: features 44..47 -> ea[4..6], bias feature 1.0
        // (load unconditionally, select with cndmask: no divergent branch)
        const float e4 = erow[4], e5 = erow[5], e6 = erow[6];
        v2f a11;
        a11.x = half ? e6   : e4;
        a11.y = half ? 1.0f : e5;
        d = __builtin_amdgcn_wmma_f32_16x16x4_f32(false, a11, false, B[11],
                                                  (short)0, d, false, false);

        // D[m][n] identical across n. C/D layout: VGPR r holds M=r (lanes 0-15)
        // and M=8+r (lanes 16-31). Lane 0 scatters edges 0..7, lane 16 edges
        // 8..15 — one exec-masked region, ids via two aligned ds_load_b128.
        if (m == 0) {
            const int base = w * 16 + 8 * half;          // 32B aligned
            const int4 ga = *(const int4*)&gbuf[base];
            const int4 gb = *(const int4*)&gbuf[base + 4];
            atomicAdd(&acc[ga.x], d[0]);
            atomicAdd(&acc[ga.y], d[1]);
            atomicAdd(&acc[ga.z], d[2]);
            atomicAdd(&acc[ga.w], d[3]);
            atomicAdd(&acc[gb.x], d[4]);
            atomicAdd(&acc[gb.y], d[5]);
            atomicAdd(&acc[gb.z], d[6]);
            atomicAdd(&acc[gb.w], d[7]);
        }
    }

    __syncthreads();
    // One global float-atomic flush per block; skip untouched graphs.
    for (int i = tid; i < NUM_GRAPHS; i += 256) {
        float p = acc[i];
        if (p != 0.f) atomicAdd(&out[i], p);
    }
}

// ---------------------------------------------------------------------------
extern "C" void kernel_launch(void* const* d_in, const int* in_sizes, int n_in,
                              void* d_out, int out_size, void* d_ws, size_t ws_size,
                              hipStream_t stream) {
    (void)in_sizes; (void)n_in; (void)out_size; (void)ws_size;
    const float* x     = (const float*)d_in[0];
    const float* ea    = (const float*)d_in[1];
    const int*   eidx  = (const int*)  d_in[2];   // JAX x64 off -> int32
    const int*   batch = (const int*)  d_in[3];
    const float* Wmsg  = (const float*)d_in[4];
    const float* bmsg  = (const float*)d_in[5];
    const float* W1    = (const float*)d_in[6];
    const float* b1    = (const float*)d_in[7];
    const float* W2    = (const float*)d_in[8];
    const float* b2    = (const float*)d_in[9];
    float* out = (float*)d_out;
    float* vws = (float*)d_ws;                    // 49 floats: v[0..47], c

    prep_kernel<<<1, 512, 0, stream>>>(Wmsg, bmsg, W1, b1, W2, b2, vws);
    init_kernel<<<(NUM_GRAPHS + 255) / 256, 256, 0, stream>>>(out, vws);
    edge_kernel<<<512, 256, 0, stream>>>(x, ea, eidx, batch, vws, out);
}